// MoE_69363721830917
// MI455X (gfx1250) — compile-verified
//
#include <hip/hip_runtime.h>
#include <hip/hip_bf16.h>
#include <math.h>
#include <stdint.h>

// ---------------- problem constants ----------------
#define D_MODEL 1024
#define D_FF    4096
#define D_FF_R  2048
#define E_R     15
#define TOPK    3
#define N_TOK   4096   // B*T = 2*2048

// ---------------- tiling ----------------
#define BM 128          // block rows   (4 waves of 32 rows)
#define BN 128          // block cols   (2 waves of 64 cols)
#define BK 32           // K chunk
#define XS_STRIDE 36    // 32 + 4 pad dwords (TDM pad codes: interval=4, amount=3)
#define WS_PITCH  288   // K-pair interleaved row: 256 dwords + 32 pad (==32 mod 64 banks)

typedef __attribute__((ext_vector_type(2))) float v2f;
typedef __attribute__((ext_vector_type(4))) float v4f;
typedef __attribute__((ext_vector_type(8))) float v8f;
typedef __attribute__((ext_vector_type(4))) unsigned int v4u;
typedef __attribute__((ext_vector_type(4))) int v4i;
typedef __attribute__((ext_vector_type(8))) int v8i;

// ------------------------------------------------------------------
// Issue one TDM 2D tile load: BK x BM tile of a [*, Kdim] row-major
// tensor into LDS at lds_lo, padded 4 dwords every 32 -> pitch 36.
// rows_rem bounds dim1 so trailing rows zero-fill (masked later anyway).
// ------------------------------------------------------------------
__device__ __forceinline__ void tdm_issue_tile_a(const float* gsrc, unsigned lds_lo,
                                                 int Kdim, int rows_rem)
{
    unsigned long long ga = (unsigned long long)(size_t)gsrc;
    v4u g0;
    g0.x = 1u;                                      // count=1, load, user mode
    g0.y = lds_lo;                                  // lds_addr (bytes)
    g0.z = (unsigned)ga;                            // global_addr[31:0]
    g0.w = (unsigned)((ga >> 32) & 0x1FFFFFFull)    // global_addr[56:32]
         | (2u << 30);                              // type = 2 (image)
    v8i g1;
    unsigned td0 = (unsigned)Kdim;                  // tensor_dim0 (elements)
    unsigned td1 = (unsigned)rows_rem;              // tensor_dim1 (rows remaining)
    g1.s0 = (2 << 16)                               // data_size = 4B
          | (1 << 20)                               // pad_enable
          | (4 << 22)                               // pad_interval: every 32 dwords
          | (3 << 25);                              // pad_amount: 4 dwords
    g1.s1 = (int)((td0 & 0xFFFFu) << 16);           // tensor_dim0 lo16 @ bits[63:48]
    g1.s2 = (int)((td0 >> 16) | ((td1 & 0xFFFFu) << 16));
    g1.s3 = (int)((td1 >> 16) | ((unsigned)BK << 16)); // tile_dim0 = BK
    g1.s4 = BM;                                     // tile_dim1 = BM
    g1.s5 = Kdim;                                   // tensor_dim0_stride lo32
    g1.s6 = 0;
    g1.s7 = 0;
    v4i z4 = {0, 0, 0, 0};
#if __clang_major__ >= 23
    v8i z8 = {0, 0, 0, 0, 0, 0, 0, 0};
    __builtin_amdgcn_tensor_load_to_lds(g0, g1, z4, z4, z8, 0);
#else
    __builtin_amdgcn_tensor_load_to_lds(g0, g1, z4, z4, 0);
#endif
}

// ------------------------------------------------------------------
// WMMA mainloop: BMxBN tile of A(rows, optionally gathered) * B[K, ldb].
// 8 waves, each owning a 32x64 strip -> 8 f32 16x16 accumulators.
// A tile staged by TDM when identity-indexed, manual gather otherwise.
// B tile staged K-pair interleaved so B fragments are single ds_load_b64.
// ------------------------------------------------------------------
__device__ __forceinline__ void gemm_tile_mainloop(
    const float* __restrict__ A, const int* __restrict__ row_idx,
    int rows, int K, const float* __restrict__ Bm, int ldb,
    int mbase, int f0, float* __restrict__ Xs, float* __restrict__ Ws,
    v8f acc[8])
{
    const int tid  = threadIdx.x;
    const int lane = tid & 31;
    const int wid  = tid >> 5;
    const int wm   = (wid & 3) << 5;   // wave row offset: 0,32,64,96
    const int wn   = (wid >> 2) << 6;  // wave col offset: 0,64
    const int half = lane >> 4;
    const int l15  = lane & 15;
    const bool use_tdm = (row_idx == nullptr);
    const unsigned lds_lo = (unsigned)(size_t)(void*)Xs; // flat lo32 == LDS byte offset

    for (int k0 = 0; k0 < K; k0 += BK) {
        // ---- A tile (BM x BK) ----
        if (use_tdm) {
            if (tid < 32)
                tdm_issue_tile_a(A + (long)mbase * K + k0, lds_lo, K, rows - mbase);
        } else {
            #pragma unroll
            for (int it = 0; it < 4; ++it) {
                int lin = tid + it * 256;          // 0..1023 : 128 rows * 8 float4
                int r   = lin >> 3;
                int c   = (lin & 7) << 2;
                int gr  = mbase + r;
                int src = gr < rows ? gr : rows - 1;
                int tok = row_idx[src];
                v4f v   = *(const v4f*)(A + (long)tok * K + k0 + c);
                *(v4f*)(Xs + r * XS_STRIDE + c) = v;
            }
        }
        // ---- B tile (BK x BN), K-pair interleaved: Ws[p][n][k&1] ----
        #pragma unroll
        for (int it = 0; it < 2; ++it) {
            int lin = tid + it * 256;              // 0..511 : 16 pairs * 32 col-groups
            int p   = lin >> 5;
            int c   = (lin & 31) << 2;
            const float* rp = Bm + (long)(k0 + (p << 1)) * ldb + f0 + c;
            v4f va = *(const v4f*)rp;              // row 2p
            v4f vb = *(const v4f*)(rp + ldb);      // row 2p+1
            v4f lo = {va.x, vb.x, va.y, vb.y};
            v4f hi = {va.z, vb.z, va.w, vb.w};
            float* dst = Ws + p * WS_PITCH + (c << 1);
            *(v4f*)dst       = lo;
            *(v4f*)(dst + 4) = hi;
        }
        // prefetch next B K-tile (global_prefetch_b8)
        {
            int pr = k0 + BK + (tid >> 3);
            int pc = f0 + ((tid & 7) << 4);
            if (pr < K) __builtin_prefetch(Bm + (long)pr * ldb + pc, 0, 1);
        }
        if (use_tdm && tid < 32) __builtin_amdgcn_s_wait_tensorcnt((short)0);
        __syncthreads();

        #pragma unroll
        for (int kk = 0; kk < BK; kk += 4) {
            // A fragments: lanes 0-15 {K=kk,kk+1}, lanes 16-31 {K=kk+2,kk+3}
            v2f a0 = *(const v2f*)(Xs + (wm + l15) * XS_STRIDE + kk + (half << 1));
            v2f a1 = *(const v2f*)(Xs + (wm + 16 + l15) * XS_STRIDE + kk + (half << 1));
            #pragma unroll
            for (int j = 0; j < 4; ++j) {
                // B fragment: contiguous pair -> one ds_load_b64
                v2f b = *(const v2f*)(Ws + ((kk >> 1) + half) * WS_PITCH
                                      + ((wn + (j << 4) + l15) << 1));
                acc[j]     = __builtin_amdgcn_wmma_f32_16x16x4_f32(
                    false, a0, false, b, (short)0, acc[j],     false, false);
                acc[4 + j] = __builtin_amdgcn_wmma_f32_16x16x4_f32(
                    false, a1, false, b, (short)0, acc[4 + j], false, false);
            }
        }
        __syncthreads();
    }
}

// ------------------------------------------------------------------
// GEMM1: H[rows, F] = gelu( X(gathered)[rows, D] * W1[D, F] + b1 )
// ------------------------------------------------------------------
__global__ __launch_bounds__(256) void MoE_gemm1(
    const float* __restrict__ X, const float* __restrict__ W1,
    const float* __restrict__ b1, float* __restrict__ H,
    const int* __restrict__ row_idx, const int* __restrict__ cnt_ptr,
    int fixed_rows, int K, int F)
{
    __shared__ __align__(16) float Xs[BM * XS_STRIDE];
    __shared__ __align__(16) float Ws[(BK / 2) * WS_PITCH];
    const int rows = cnt_ptr ? *cnt_ptr : fixed_rows;
    const int mbase = blockIdx.y * BM;
    if (mbase >= rows) return;
    const int f0 = blockIdx.x * BN;

    v8f zero = {0.f,0.f,0.f,0.f,0.f,0.f,0.f,0.f};
    v8f acc[8] = {zero, zero, zero, zero, zero, zero, zero, zero};
    gemm_tile_mainloop(X, row_idx, rows, K, W1, F, mbase, f0, Xs, Ws, acc);

    const int tid = threadIdx.x, lane = tid & 31, wid = tid >> 5;
    const int wm = (wid & 3) << 5, wn = (wid >> 2) << 6;
    const int half = lane >> 4, l15 = lane & 15;
    #pragma unroll
    for (int j = 0; j < 4; ++j) {
        int col  = f0 + wn + (j << 4) + l15;
        float bb = b1[col];
        #pragma unroll
        for (int mi = 0; mi < 2; ++mi) {
            #pragma unroll
            for (int i = 0; i < 8; ++i) {
                int gr = mbase + wm + (mi << 4) + i + (half << 3);
                if (gr < rows) {
                    float v = acc[mi * 4 + j][i] + bb;
                    v = 0.5f * v * (1.0f + erff(v * 0.70710678118654752f)); // exact gelu
                    H[(long)gr * F + col] = v;
                }
            }
        }
    }
}

// ------------------------------------------------------------------
// GEMM2: acc = H[rows, K] * W2[K, D] ;  epilogue:
//   mode 0 (shared): out[r]   = acc + b2 + x[r]       (residual fused)
//   mode 1 (router): out[tok] += gate * (acc + b2)    (scatter, per-expert)
// ------------------------------------------------------------------
__global__ __launch_bounds__(256) void MoE_gemm2(
    const float* __restrict__ H, const float* __restrict__ W2,
    const float* __restrict__ b2, const float* __restrict__ Xres,
    float* __restrict__ out,
    const int* __restrict__ row_idx, const float* __restrict__ row_gate,
    const int* __restrict__ cnt_ptr, int fixed_rows, int K, int mode)
{
    __shared__ __align__(16) float Xs[BM * XS_STRIDE];
    __shared__ __align__(16) float Ws[(BK / 2) * WS_PITCH];
    const int rows = cnt_ptr ? *cnt_ptr : fixed_rows;
    const int mbase = blockIdx.y * BM;
    if (mbase >= rows) return;
    const int f0 = blockIdx.x * BN;

    v8f zero = {0.f,0.f,0.f,0.f,0.f,0.f,0.f,0.f};
    v8f acc[8] = {zero, zero, zero, zero, zero, zero, zero, zero};
    gemm_tile_mainloop(H, nullptr, rows, K, W2, D_MODEL, mbase, f0, Xs, Ws, acc);

    const int tid = threadIdx.x, lane = tid & 31, wid = tid >> 5;
    const int wm = (wid & 3) << 5, wn = (wid >> 2) << 6;
    const int half = lane >> 4, l15 = lane & 15;
    #pragma unroll
    for (int j = 0; j < 4; ++j) {
        int col  = f0 + wn + (j << 4) + l15;
        float bb = b2[col];
        #pragma unroll
        for (int mi = 0; mi < 2; ++mi) {
            #pragma unroll
            for (int i = 0; i < 8; ++i) {
                int gr = mbase + wm + (mi << 4) + i + (half << 3);
                if (gr < rows) {
                    float v = acc[mi * 4 + j][i] + bb;
                    if (mode == 0) {
                        long o = (long)gr * D_MODEL + col;
                        out[o] = v + Xres[o];
                    } else {
                        int tok = row_idx[gr];
                        float g = row_gate[gr];
                        long o = (long)tok * D_MODEL + col;
                        out[o] += g * v;   // unique (tok,col) within this launch
                    }
                }
            }
        }
    }
}

// ------------------------------------------------------------------
// Gating: one wave32 per token. logits = x@gW + gb, softmax, top-3
// (strict > scan => lowest index wins ties, matching lax.top_k).
// ------------------------------------------------------------------
__global__ __launch_bounds__(256) void MoE_gate(
    const float* __restrict__ x, const float* __restrict__ gW,
    const float* __restrict__ gb, int* __restrict__ top_idx,
    float* __restrict__ top_val)
{
    int t    = (blockIdx.x * blockDim.x + threadIdx.x) >> 5;
    int lane = threadIdx.x & 31;
    if (t >= N_TOK) return;
    const float* xr = x + (long)t * D_MODEL;
    float p[E_R];
    #pragma unroll
    for (int e = 0; e < E_R; ++e) p[e] = 0.f;
    for (int d = lane; d < D_MODEL; d += 32) {
        float xv = xr[d];
        const float* g = gW + d * E_R;
        #pragma unroll
        for (int e = 0; e < E_R; ++e) p[e] += xv * g[e];
    }
    #pragma unroll
    for (int e = 0; e < E_R; ++e) {
        float v = p[e];
        for (int off = 16; off > 0; off >>= 1) v += __shfl_down(v, off, 32);
        p[e] = v;
    }
    if (lane == 0) {
        float w[E_R], mx = -1e30f;
        #pragma unroll
        for (int e = 0; e < E_R; ++e) { w[e] = p[e] + gb[e]; mx = fmaxf(mx, w[e]); }
        float s = 0.f;
        #pragma unroll
        for (int e = 0; e < E_R; ++e) { w[e] = expf(w[e] - mx); s += w[e]; }
        float inv = 1.f / s;
        #pragma unroll
        for (int e = 0; e < E_R; ++e) w[e] *= inv;
        #pragma unroll
        for (int k = 0; k < TOPK; ++k) {
            int bi = 0; float bv = w[0];
            #pragma unroll
            for (int e = 1; e < E_R; ++e) if (w[e] > bv) { bv = w[e]; bi = e; }
            top_idx[t * TOPK + k] = bi;
            top_val[t * TOPK + k] = bv;
            w[bi] = -1.f;
        }
    }
}

// ------------------------------------------------------------------
// Build per-expert gathered token lists. List order is nondeterministic
// but the final output is order-invariant (unique (token,expert) slots,
// per-expert launches serialized on the stream).
// ------------------------------------------------------------------
__global__ void MoE_build(const int* __restrict__ top_idx,
                          const float* __restrict__ top_val,
                          int* __restrict__ cnt,
                          int* __restrict__ rows_tok,
                          float* __restrict__ rows_gate)
{
    int i = blockIdx.x * blockDim.x + threadIdx.x;
    if (i >= N_TOK * TOPK) return;
    int e = top_idx[i];
    int pos = atomicAdd(&cnt[e], 1);
    rows_tok[e * N_TOK + pos]  = i / TOPK;
    rows_gate[e * N_TOK + pos] = top_val[i];
}

__global__ void MoE_zero(int* __restrict__ p, int n)
{
    int i = blockIdx.x * blockDim.x + threadIdx.x;
    if (i < n) p[i] = 0;
}

// ------------------------------------------------------------------
extern "C" void kernel_launch(void* const* d_in, const int* in_sizes, int n_in,
                              void* d_out, int out_size, void* d_ws, size_t ws_size,
                              hipStream_t stream)
{
    (void)in_sizes; (void)n_in; (void)out_size; (void)ws_size;
    const float* x   = (const float*)d_in[0];
    const float* sW1 = (const float*)d_in[1];
    const float* sb1 = (const float*)d_in[2];
    const float* sW2 = (const float*)d_in[3];
    const float* sb2 = (const float*)d_in[4];
    const float* rW1 = (const float*)d_in[5];
    const float* rb1 = (const float*)d_in[6];
    const float* rW2 = (const float*)d_in[7];
    const float* rb2 = (const float*)d_in[8];
    const float* gW  = (const float*)d_in[9];
    const float* gb  = (const float*)d_in[10];
    float* out = (float*)d_out;

    // workspace layout (~64.6 MB)
    float* H        = (float*)d_ws;                      // N_TOK * D_FF (reused)
    int*   top_idx  = (int*)(H + (size_t)N_TOK * D_FF);  // N_TOK*TOPK
    float* top_val  = (float*)(top_idx + N_TOK * TOPK);  // N_TOK*TOPK
    int*   cnt      = (int*)(top_val + N_TOK * TOPK);    // E_R (padded 64)
    int*   rows_tok = cnt + 64;                          // E_R * N_TOK
    float* rows_gate= (float*)(rows_tok + E_R * N_TOK);  // E_R * N_TOK

    MoE_zero <<<1, 64, 0, stream>>>(cnt, E_R);
    MoE_gate <<<N_TOK / 8, 256, 0, stream>>>(x, gW, gb, top_idx, top_val);
    MoE_build<<<(N_TOK * TOPK + 255) / 256, 256, 0, stream>>>(
        top_idx, top_val, cnt, rows_tok, rows_gate);

    // shared expert: dense over all tokens (TDM-staged A), residual fused
    MoE_gemm1<<<dim3(D_FF / BN, N_TOK / BM), 256, 0, stream>>>(
        x, sW1, sb1, H, nullptr, nullptr, N_TOK, D_MODEL, D_FF);
    MoE_gemm2<<<dim3(D_MODEL / BN, N_TOK / BM), 256, 0, stream>>>(
        H, sW2, sb2, x, out, nullptr, nullptr, nullptr, N_TOK, D_FF, 0);

    // router experts: gathered tokens only (5x less matrix work than dense).
    // Grid covers worst case; blocks past cnt[e] early-exit.
    for (int e = 0; e < E_R; ++e) {
        MoE_gemm1<<<dim3(D_FF_R / BN, N_TOK / BM), 256, 0, stream>>>(
            x, rW1 + (size_t)e * D_MODEL * D_FF_R, rb1 + (size_t)e * D_FF_R, H,
            rows_tok + e * N_TOK, cnt + e, 0, D_MODEL, D_FF_R);
        MoE_gemm2<<<dim3(D_MODEL / BN, N_TOK / BM), 256, 0, stream>>>(
            H, rW2 + (size_t)e * D_FF_R * D_MODEL, rb2 + (size_t)e * D_MODEL,
            nullptr, out, rows_tok + e * N_TOK, rows_gate + e * N_TOK,
            cnt + e, 0, D_FF_R, 1);
    }
}